// Model_84542136254884
// MI455X (gfx1250) — compile-verified
//
#include <hip/hip_runtime.h>
#include <math.h>

typedef __attribute__((ext_vector_type(2))) float    v2f;
typedef __attribute__((ext_vector_type(8))) float    v8f;
typedef __attribute__((ext_vector_type(4))) unsigned v4u;
typedef __attribute__((ext_vector_type(8))) int      v8i;
typedef __attribute__((ext_vector_type(4))) int      v4i;

#define H       256
#define NUM_V   512
#define NUM_F   1024
#define NGROUP  (NUM_F / 16)        // 64 face groups of 16
#define NPIX    (H * H)             // 65536
#define NTILE   (NPIX / 16)         // 4096 pixel tiles
#define WAVES_PER_BLOCK 8
#define TAN_T     0.57735026919f    // tan(30 deg)
#define INV_SIGMA 1.0e4f            // 1/SIGMA
#define SP_CLAMP  13.815510558f     // -log(1e-6)  (matches prob clip)
#define BIMG_FLOATS (NGROUP * 3 * 64)   // 12288 floats = 48 KB
#define WS_BIMG_OFF 64              // floats; ws[0] = loss accumulator

// ---------------------------------------------------------------------------
// Kernel 0: zero the scalar loss accumulator in workspace
// ---------------------------------------------------------------------------
__global__ void k_zero(float* ws) {
    if (threadIdx.x == 0) ws[0] = 0.0f;
}

// ---------------------------------------------------------------------------
// Kernel 1: per-face setup. Transform+project vertices, build edge-coefficient
// table directly in WMMA B-operand lane layout, with the two B VGPRs
// interleaved per lane so the main loop does one ds_load_b64 per edge:
//   pair index pe = g*3+e, lane l: bimg[pe*64 + 2*l] = B_vgpr0[l],
//                                  bimg[pe*64 + 2*l + 1] = B_vgpr1[l]
//   B_vgpr0: lanes 0-15 = Acoef (K=0 row), lanes 16-31 = Ccoef (K=2 row)
//   B_vgpr1: lanes 0-15 = Bcoef (K=1 row), lanes 16-31 = 0     (K=3 row)
// Invalid faces (any z <= 0.001) get degenerate coefficients that force
// dist = -1 -> softplus(-1e4) = 0 contribution.
// ---------------------------------------------------------------------------
__global__ void k_setup(const float* __restrict__ verts,
                        const int*   __restrict__ faces,
                        const float* __restrict__ cam,
                        float*       __restrict__ bimg) {
    int f = blockIdx.x * blockDim.x + threadIdx.x;
    if (f >= NUM_F) return;

    const float ex = cam[0], ey = cam[1], ez = cam[2];
    // camera basis (same per thread; cheap)
    float inv = 1.0f / (sqrtf(ex * ex + ey * ey + ez * ez) + 1e-8f);
    float zx = -ex * inv, zy = -ey * inv, zz = -ez * inv;
    // x_axis = normalize(cross(up=(0,1,0), z)) = normalize(zz, 0, -zx)
    float cx = zz, cy = 0.0f, cz = -zx;
    inv = 1.0f / (sqrtf(cx * cx + cy * cy + cz * cz) + 1e-8f);
    float xx = cx * inv, xy = cy * inv, xz = cz * inv;
    // y_axis = normalize(cross(z, x))
    float yx = zy * xz - zz * xy;
    float yy = zz * xx - zx * xz;
    float yz = zx * xy - zy * xx;
    inv = 1.0f / (sqrtf(yx * yx + yy * yy + yz * yz) + 1e-8f);
    yx *= inv; yy *= inv; yz *= inv;

    float px[3], py[3];
    bool valid = true;
    #pragma unroll
    for (int k = 0; k < 3; ++k) {
        int vi = faces[f * 3 + k];
        float wx = verts[vi * 3 + 0] - ex;
        float wy = verts[vi * 3 + 1] - ey;
        float wz = verts[vi * 3 + 2] - ez;
        float vx = xx * wx + xy * wy + xz * wz;
        float vy = yx * wx + yy * wy + yz * wz;
        float vz = zx * wx + zy * wy + zz * wz;
        valid = valid && (vz > 0.001f);
        float d = 1.0f / (vz * TAN_T + 1e-8f);
        px[k] = vx * d;
        py[k] = vy * d;
    }

    int g = f >> 4, n = f & 15;
    #pragma unroll
    for (int e = 0; e < 3; ++e) {
        int a = e, b = (e + 1 == 3) ? 0 : e + 1;   // edges (0,1),(1,2),(2,0)
        float Ac, Bc, Cc;
        if (valid) {
            float evx = px[b] - px[a], evy = py[b] - py[a];
            float il = 1.0f / (sqrtf(evx * evx + evy * evy) + 1e-8f);
            // cross = evx*(qy - ay) - evy*(qx - ax)
            Ac = -evy * il;
            Bc =  evx * il;
            Cc = (evy * px[a] - evx * py[a]) * il;
        } else {
            // d0=d1=-1, d2=+1 -> dmin=-1, dmax=1, dist=-1 -> zero prob
            Ac = 0.0f; Bc = 0.0f; Cc = (e == 2) ? 1.0f : -1.0f;
        }
        float* base = bimg + (g * 3 + e) * 64;
        base[2 * n]            = Ac;    // lane n,   B vgpr0 (K=0)
        base[2 * n + 1]        = Bc;    // lane n,   B vgpr1 (K=1)
        base[32 + 2 * n]       = Cc;    // lane 16+n, B vgpr0 (K=2)
        base[32 + 2 * n + 1]   = 0.0f;  // lane 16+n, B vgpr1 (K=3)
    }
}

// ---------------------------------------------------------------------------
// Kernel 2: main. One wave per 16-pixel tile. Coefficient table staged into
// LDS once per block by the Tensor Data Mover (async DMA, TENSORcnt), then
// each face group costs 3 ds_load_b64 + 3 V_WMMA_F32_16X16X4_F32 + VALU
// min/max + clamped-softplus epilogue.
// ---------------------------------------------------------------------------
__global__ void __launch_bounds__(WAVES_PER_BLOCK * 32)
k_main(const float* __restrict__ bimg,
       const float* __restrict__ image_ref,
       float*       __restrict__ loss_acc) {
    __shared__ float lds_b[BIMG_FLOATS];   // 48 KB

    const int lane = threadIdx.x & 31;
    const int wave = threadIdx.x >> 5;
    const int tile = blockIdx.x * WAVES_PER_BLOCK + wave;
    const int m    = lane & 15;
    const int pix0 = tile * 16;

    // ---- TDM: DMA the 48 KB coefficient image into LDS (wave 0 only) ----
    if (wave == 0) {
        // LDS aperture: low 32 bits of the generic shared address are the
        // LDS byte offset (ISA 10.2).
        unsigned long long ga = (unsigned long long)bimg;
        unsigned lds_off = (unsigned)(unsigned long long)(void*)&lds_b[0];

        v4u g0;
        g0[0] = 1u;                                   // count=1, user mode
        g0[1] = lds_off;                              // lds_addr (bytes)
        g0[2] = (unsigned)(ga & 0xFFFFFFFFull);       // global_addr[31:0]
        g0[3] = (unsigned)((ga >> 32) & 0x01FFFFFFull)// global_addr[56:32]
                | 0x80000000u;                        // type=2 ("image")

        v8i g1;
        g1[0] = 0x20000;              // wg_mask=0, data_size=2 (4 bytes)
        g1[1] = (int)0x30000000u;     // tensor_dim0[15:0]=0x3000 in [31:16]
        g1[2] = (int)0x00010000u;     // tensor_dim0[31:16]=0, tensor_dim1=1
        g1[3] = (int)0x30000000u;     // tensor_dim1 hi=0, tile_dim0=0x3000
        g1[4] = 1;                    // tile_dim1=1, tile_dim2=0
        g1[5] = 0x3000;               // tensor_dim0_stride = 12288
        g1[6] = (int)0x30000000u;     // tensor_dim1_stride[15:0]=0x3000
        g1[7] = 0;

        v4i g2 = {0, 0, 0, 0};
        v4i g3 = {0, 0, 0, 0};
        v8i g4 = {0, 0, 0, 0, 0, 0, 0, 0};
        __builtin_amdgcn_tensor_load_to_lds(g0, g1, g2, g3, g4, 0);
        __builtin_amdgcn_s_wait_tensorcnt(0);
    }
    __syncthreads();

    // A operand: rows = pixels, K = (qx, qy, 1, 0)
    int p   = pix0 + m;
    int col = p & (H - 1);
    int row = p >> 8;
    float qx = (col + 0.5f) * (2.0f / H) - 1.0f;
    float qy = 1.0f - (row + 0.5f) * (2.0f / H);
    v2f A;
    A.x = (lane < 16) ? qx : 1.0f;   // A vgpr0: K=0 (lanes 0-15) / K=2
    A.y = (lane < 16) ? qy : 0.0f;   // A vgpr1: K=1 / K=3

    v8f acc = {0.f, 0.f, 0.f, 0.f, 0.f, 0.f, 0.f, 0.f};  // sum log1p(-p)

    for (int g = 0; g < NGROUP; ++g) {
        const float2* basep = (const float2*)(lds_b + g * 3 * 64);
        float2 p0 = basep[lane];          // edge 0 pair  -> ds_load_b64
        float2 p1 = basep[32 + lane];     // edge 1 pair
        float2 p2 = basep[64 + lane];     // edge 2 pair
        v2f B0, B1, B2;
        B0.x = p0.x; B0.y = p0.y;
        B1.x = p1.x; B1.y = p1.y;
        B2.x = p2.x; B2.y = p2.y;

        v8f z = {0.f, 0.f, 0.f, 0.f, 0.f, 0.f, 0.f, 0.f};
        v8f d0 = __builtin_amdgcn_wmma_f32_16x16x4_f32(false, A, false, B0,
                                                       (short)0, z, false, false);
        v8f d1 = __builtin_amdgcn_wmma_f32_16x16x4_f32(false, A, false, B1,
                                                       (short)0, z, false, false);
        v8f d2 = __builtin_amdgcn_wmma_f32_16x16x4_f32(false, A, false, B2,
                                                       (short)0, z, false, false);
        #pragma unroll
        for (int j = 0; j < 8; ++j) {
            float e0 = d0[j], e1 = d1[j], e2 = d2[j];
            float dmin = fminf(fminf(e0, e1), e2);
            float dmax = fmaxf(fmaxf(e0, e1), e2);
            float dist = fmaxf(dmin, -dmax);
            float y  = dist * fabsf(dist) * INV_SIGMA;
            // log1p(-sigmoid(y)) = -softplus(y); prob clip -> clamp SP_CLAMP
            float sp = fmaxf(y, 0.0f) + __logf(1.0f + __expf(-fabsf(y)));
            sp = fminf(sp, SP_CLAMP);
            acc[j] -= sp;
        }
    }

    // Reduce over the 16 face-columns: butterfly inside each 16-lane half.
    #pragma unroll
    for (int j = 0; j < 8; ++j) {
        float a = acc[j];
        a += __shfl_xor(a, 1, 32);
        a += __shfl_xor(a, 2, 32);
        a += __shfl_xor(a, 4, 32);
        a += __shfl_xor(a, 8, 32);
        acc[j] = a;
    }

    // lanes 0-15 hold pixels M=j (pix0+j), lanes 16-31 hold M=j+8.
    if (lane == 0 || lane == 16) {
        int base_px = pix0 + ((lane == 16) ? 8 : 0);
        float lsum = 0.0f;
        #pragma unroll
        for (int j = 0; j < 8; ++j) {
            float alpha = 1.0f - __expf(acc[j]);
            float diff  = image_ref[base_px + j] - alpha;
            lsum += diff * diff;
        }
        atomicAdd(loss_acc, lsum);
    }
}

// ---------------------------------------------------------------------------
// Kernel 3: finalize with distance penalty
// ---------------------------------------------------------------------------
__global__ void k_final(const float* __restrict__ loss_acc,
                        const float* __restrict__ cam,
                        float* __restrict__ out) {
    if (threadIdx.x == 0 && blockIdx.x == 0) {
        float ex = cam[0], ey = cam[1], ez = cam[2];
        float dist = sqrtf(ex * ex + ey * ey + ez * ez);
        float pen = fmaxf(0.0f, 6.0f - dist);
        out[0] = loss_acc[0] * (1.0f + pen);
    }
}

// ---------------------------------------------------------------------------
extern "C" void kernel_launch(void* const* d_in, const int* in_sizes, int n_in,
                              void* d_out, int out_size, void* d_ws, size_t ws_size,
                              hipStream_t stream) {
    (void)in_sizes; (void)n_in; (void)out_size; (void)ws_size;
    const float* verts     = (const float*)d_in[0];  // [1,512,3] f32
    const int*   faces     = (const int*)  d_in[1];  // [1,1024,3] i32
    const float* image_ref = (const float*)d_in[2];  // [256,256] f32
    const float* cam       = (const float*)d_in[3];  // [3] f32

    float* ws   = (float*)d_ws;
    float* lacc = ws;                 // ws[0]
    float* bimg = ws + WS_BIMG_OFF;   // BIMG_FLOATS floats = 48 KB

    k_zero<<<1, 32, 0, stream>>>(lacc);
    k_setup<<<(NUM_F + 255) / 256, 256, 0, stream>>>(verts, faces, cam, bimg);
    k_main<<<NTILE / WAVES_PER_BLOCK, WAVES_PER_BLOCK * 32, 0, stream>>>(
        bimg, image_ref, lacc);
    k_final<<<1, 32, 0, stream>>>(lacc, cam, (float*)d_out);
}